// _Shift_13615046329050
// MI455X (gfx1250) — compile-verified
//
#include <hip/hip_runtime.h>
#include <cstdint>

// Shift op: out[n,c,h,w] = x[n,c,h+di_c,w+dj_c] (zero padded), (di,dj) decoded
// from the one-hot depthwise kernel. Pure data movement: ~1.07 GB @ 23.3 TB/s
// => ~46us floor, 0 FLOPs => WMMA inapplicable. CDNA5 async DMA path:
//   HBM --global_load_async_to_lds_b128--> LDS --global_store_async_from_lds_b32--> HBM
// Column shift resolved by guard-zero columns in the LDS row buffer; the
// offset: immediate advances BOTH the global and LDS addresses (ISA 08 §4.4),
// so each row needs exactly one address setup.

#define N_B   4
#define N_C   128
#define N_H   512
#define N_W   512
#define ROWS  16            // rows per block
#define RPW   2             // rows per wave (8 wave32 * 2 = 16)
#define LROW  520           // [3]=left guard, [4..515]=cols 0..511, [516]=right guard

__global__ __launch_bounds__(128)
void decode_shifts(const float* __restrict__ kern, int* __restrict__ shifts) {
    int c = threadIdx.x;
    if (c < N_C) {
        int p = 4;  // default: center tap
#pragma unroll
        for (int t = 0; t < 9; ++t)
            if (kern[c * 9 + t] > 0.5f) p = t;
        shifts[c] = p;  // p = 3*i0 + j0
    }
}

// One address setup per row, literal offset per step (applies to both sides).
#define ALOAD128(l0, g0, OFS)                                                  \
    asm volatile("global_load_async_to_lds_b128 %0, %1, off offset:" #OFS      \
                 " th:TH_LOAD_NT" :: "v"(l0), "v"(g0) : "memory")
#define ASTORE32(g0, l0, OFS)                                                  \
    asm volatile("global_store_async_from_lds_b32 %0, %1, off offset:" #OFS    \
                 " th:TH_STORE_NT" :: "v"(g0), "v"(l0) : "memory")

__global__ __launch_bounds__(256)
void shift_async_kernel(const float* __restrict__ x,
                        const int* __restrict__ shifts,
                        float* __restrict__ out) {
    __shared__ __align__(16) float lds[ROWS][LROW];

    const int plane = blockIdx.x >> 5;          // 32 row-groups per 512-row plane
    const int rg    = blockIdx.x & 31;
    const int c     = plane & (N_C - 1);        // plane = n*128 + c
    const int hbase = rg * ROWS;

    const int p  = shifts[c];
    const int di = p / 3 - 1;                   // row shift in {-1,0,1}
    const int dj = p % 3 - 1;                   // col shift in {-1,0,1}

    const int wave = threadIdx.x >> 5;          // wave32
    const int lane = threadIdx.x & 31;
    const long planeOff = (long)plane * (N_H * N_W);

    // ---- Phase 1: fill this wave's LDS rows (async b128 HBM->LDS, or ds-zero)
#pragma unroll
    for (int rr = 0; rr < RPW; ++rr) {
        const int r  = wave * RPW + rr;
        const int hs = hbase + r + di;          // source row, may be -1 or 512
        if (hs < 0 || hs >= N_H) {
            // out-of-range source row -> zero data + guards (indices 3..516)
            for (int col = 3 + lane; col <= 516; col += 32)
                lds[r][col] = 0.0f;
        } else {
            // guard zeros absorb the w-1 / w+1 column reads in phase 2
            if (lane < 2) lds[r][lane == 0 ? 3 : 516] = 0.0f;
            // lane l loads cols 4l..4l+3 (+128 per step); LDS dst 16B-aligned
            const float* g0 = x + planeOff + (long)hs * N_W + 4 * lane;
            uint32_t l0 = (uint32_t)(uintptr_t)&lds[r][4 + 4 * lane];
            ALOAD128(l0, g0, 0);
            ALOAD128(l0, g0, 512);
            ALOAD128(l0, g0, 1024);
            ALOAD128(l0, g0, 1536);
        }
    }
    // Wave reads only LDS it wrote itself -> no barrier, just drain counters.
    asm volatile("s_wait_asynccnt 0x0" ::: "memory");
    asm volatile("s_wait_dscnt 0x0"    ::: "memory");

    // ---- Phase 2: async b32 LDS->HBM stores, column shift on the LDS side ----
#pragma unroll
    for (int rr = 0; rr < RPW; ++rr) {
        const int r = wave * RPW + rr;
        const int h = hbase + r;
        const float* g0 = out + planeOff + (long)h * N_W + lane;
        uint32_t l0 = (uint32_t)(uintptr_t)&lds[r][4 + dj + lane];
        ASTORE32(g0, l0, 0);    ASTORE32(g0, l0, 128);
        ASTORE32(g0, l0, 256);  ASTORE32(g0, l0, 384);
        ASTORE32(g0, l0, 512);  ASTORE32(g0, l0, 640);
        ASTORE32(g0, l0, 768);  ASTORE32(g0, l0, 896);
        ASTORE32(g0, l0, 1024); ASTORE32(g0, l0, 1152);
        ASTORE32(g0, l0, 1280); ASTORE32(g0, l0, 1408);
        ASTORE32(g0, l0, 1536); ASTORE32(g0, l0, 1664);
        ASTORE32(g0, l0, 1792); ASTORE32(g0, l0, 1920);
    }
    // S_ENDPGM performs an implicit wait-idle, draining ASYNCcnt before the
    // LDS allocation is released.
}

extern "C" void kernel_launch(void* const* d_in, const int* in_sizes, int n_in,
                              void* d_out, int out_size, void* d_ws, size_t ws_size,
                              hipStream_t stream) {
    const float* x    = (const float*)d_in[0];   // [4,128,512,512] f32
    const float* kern = (const float*)d_in[1];   // [128,1,3,3] one-hot f32
    float* out        = (float*)d_out;           // [4,128,512,512] f32
    int*   shifts     = (int*)d_ws;              // 128 ints of scratch

    decode_shifts<<<1, 128, 0, stream>>>(kern, shifts);

    const int blocks = N_B * N_C * (N_H / ROWS); // 16384
    shift_async_kernel<<<blocks, 256, 0, stream>>>(x, shifts, out);
}